// Graph_encoder_8383776162381
// MI455X (gfx1250) — compile-verified
//
#include <hip/hip_runtime.h>
#include <math.h>

// ---- problem constants (match reference) ----
#define Nn 20000
#define Ee 320000
#define Bb 32
#define Hh 4
#define Dd 128
#define EPSf 1e-5f

typedef float v2f __attribute__((ext_vector_type(2)));
typedef float v8f __attribute__((ext_vector_type(8)));

// ---------------- order-preserving float <-> uint for atomic max ----------------
__device__ __forceinline__ unsigned enc_f(float f) {
  unsigned u = __float_as_uint(f);
  return (u & 0x80000000u) ? ~u : (u | 0x80000000u);
}
__device__ __forceinline__ float dec_f(unsigned u) {
  u = (u & 0x80000000u) ? (u & 0x7FFFFFFFu) : ~u;
  return __uint_as_float(u);
}

// ---------------- generic fp32 WMMA GEMM ----------------
// C[M,Nc] (accum? += : =) act( A'[M,K] @ B[K,N] + bias ),  A' = gather ? A[gather[m]] : A[m]
// One wave32 computes one 16x16 tile via v_wmma_f32_16x16x4_f32.
// A fragment (16x4): lanes 0-15 hold M=lane, K = {0,1}; lanes 16-31 hold M=lane-16, K = {2,3}
// B fragment (4x16): lanes 0-15 hold N=lane, K = {0,1}; lanes 16-31 hold N=lane-16, K = {2,3}
// C/D (16x16): vgpr j: lanes 0-15 -> (M=j, N=lane); lanes 16-31 -> (M=j+8, N=lane-16)
__global__ void gemm_f32_wmma(const float* __restrict__ A,
                              const float* __restrict__ B,
                              const float* __restrict__ bias,
                              float* __restrict__ C,
                              const int* __restrict__ gather,
                              int M, int K, int Nc,
                              int accum, int act) {
  const int lane = threadIdx.x & 31;
  const int wave = threadIdx.x >> 5;
  const int mtile = (blockIdx.x * (blockDim.x >> 5) + wave) * 16;
  const int ntile = blockIdx.y * 16;
  if (mtile >= M) return;

  const int half = lane >> 4;   // 0: K pair {0,1}, 1: K pair {2,3}
  const int mi   = lane & 15;

  const int row  = mtile + mi;
  const int arow = gather ? gather[row] : row;
  const float* Ap = A + (long)arow * K + 2 * half;        // A row-major [M,K]
  const float* Bp = B + (long)(2 * half) * Nc + ntile + mi; // B row-major [K,N]

  v8f c = {};
  for (int k = 0; k < K; k += 4) {
    v2f a, b;
    a.x = Ap[k];
    a.y = Ap[k + 1];
    b.x = Bp[(long)k * Nc];
    b.y = Bp[(long)(k + 1) * Nc];
    c = __builtin_amdgcn_wmma_f32_16x16x4_f32(false, a, false, b,
                                              (short)0, c, false, false);
  }

#pragma unroll
  for (int j = 0; j < 8; ++j) {
    const int r   = mtile + j + 8 * half;
    const int col = ntile + mi;
    const long idx = (long)r * Nc + col;
    float v = c[j];
    if (accum) {
      C[idx] += v;              // single writer per element per pass
    } else {
      if (bias) v += bias[col];
      if (act == 1) v = fmaxf(v, 0.f);
      C[idx] = v;
    }
  }
}

// ---------------- elementwise / reduction kernels ----------------
__global__ void zero_f(float* p, long n) {
  long i = (long)blockIdx.x * blockDim.x + threadIdx.x;
  if (i < n) p[i] = 0.f;
}

__global__ void init_mx(unsigned* p, long n) {
  long i = (long)blockIdx.x * blockDim.x + threadIdx.x;
  if (i < n) p[i] = enc_f(-__builtin_inff());
}

// per-column batch stats: stats[0..127]=sum, stats[128..255]=sumsq of (x + x2 + cb)
__global__ void bn_stats(const float* __restrict__ x, const float* __restrict__ x2,
                         const float* __restrict__ cb, float* __restrict__ stats,
                         int rows) {
  const int c = threadIdx.x;          // 128 threads, one per column
  const int r0 = blockIdx.x * 128;
  const int r1 = min(r0 + 128, rows);
  float s = 0.f, ss = 0.f;
  for (int r = r0; r < r1; ++r) {
    float v = x[(long)r * Dd + c];
    if (x2) v += x2[(long)r * Dd + c];
    if (cb) v += cb[c];
    s += v; ss += v * v;
  }
  atomicAdd(&stats[c], s);
  atomicAdd(&stats[Dd + c], ss);
}

__global__ void bn_apply(const float* __restrict__ x, const float* __restrict__ x2,
                         const float* __restrict__ cb, const float* __restrict__ stats,
                         const float* __restrict__ g, const float* __restrict__ b,
                         float* __restrict__ out, int rows) {
  long i = (long)blockIdx.x * blockDim.x + threadIdx.x;
  long total = (long)rows * Dd;
  if (i >= total) return;
  int c = i & (Dd - 1);
  float m = stats[c] / (float)rows;
  float var = stats[Dd + c] / (float)rows - m * m;  // biased variance
  float rs = rsqrtf(var + EPSf);
  float v = x[i];
  if (x2) v += x2[i];
  if (cb) v += cb[c];
  out[i] = (v - m) * rs * g[c] + b[c];
}

// layernorm over 128 with residual: hv = ln(hv + res) * g + b ; one wave per row
__global__ void ln_res(float* __restrict__ hv, const float* __restrict__ res,
                       const float* __restrict__ g, const float* __restrict__ b,
                       int rows) {
  const int row = blockIdx.x * (blockDim.x >> 5) + (threadIdx.x >> 5);
  const int lane = threadIdx.x & 31;
  if (row >= rows) return;
  float x[4];
  float s = 0.f;
#pragma unroll
  for (int i = 0; i < 4; ++i) {
    long idx = (long)row * Dd + lane + 32 * i;
    x[i] = hv[idx] + res[idx];
    s += x[i];
  }
#pragma unroll
  for (int off = 16; off > 0; off >>= 1) s += __shfl_xor(s, off, 32);
  const float m = s * (1.f / 128.f);
  float vs = 0.f;
#pragma unroll
  for (int i = 0; i < 4; ++i) { float d = x[i] - m; vs += d * d; }
#pragma unroll
  for (int off = 16; off > 0; off >>= 1) vs += __shfl_xor(vs, off, 32);
  const float rs = rsqrtf(vs * (1.f / 128.f) + EPSf);
#pragma unroll
  for (int i = 0; i < 4; ++i) {
    int c = lane + 32 * i;
    hv[(long)row * Dd + c] = (x[i] - m) * rs * g[c] + b[c];
  }
}

// ---------------- attention ----------------
__global__ void attn_logits(const float* __restrict__ qn, const float* __restrict__ kn,
                            const float* __restrict__ eatt,
                            const int* __restrict__ src, const int* __restrict__ dst,
                            float* __restrict__ logits, unsigned* __restrict__ mx) {
  long t = (long)blockIdx.x * blockDim.x + threadIdx.x;
  if (t >= (long)Ee * Hh) return;
  const int e = (int)(t >> 2), h = (int)(t & 3);
  const int s = src[e], d = dst[e];
  const float* qp = qn + (long)d * Dd + h * 32;
  const float* kp = kn + (long)s * Dd + h * 32;
  const float* ep = eatt + (long)e * Dd + h * 32;
  float acc = 0.f;
#pragma unroll
  for (int i = 0; i < 32; i += 4) {
    float4 q = *(const float4*)(qp + i);
    float4 k = *(const float4*)(kp + i);
    float4 x = *(const float4*)(ep + i);
    acc += q.x * (k.x + x.x) + q.y * (k.y + x.y) + q.z * (k.z + x.z) + q.w * (k.w + x.w);
  }
  acc *= 0.17677669529663687f;  // 1/sqrt(32)
  logits[t] = acc;
  atomicMax(&mx[(long)d * Hh + h], enc_f(acc));
}

__global__ void attn_exden(const float* __restrict__ logits, const int* __restrict__ dst,
                           const unsigned* __restrict__ mx, float* __restrict__ ex,
                           float* __restrict__ den) {
  long t = (long)blockIdx.x * blockDim.x + threadIdx.x;
  if (t >= (long)Ee * Hh) return;
  const int e = (int)(t >> 2), h = (int)(t & 3);
  const int d = dst[e];
  float v = expf(logits[t] - dec_f(mx[(long)d * Hh + h]));
  ex[t] = v;
  atomicAdd(&den[(long)d * Hh + h], v);
}

__global__ void attn_accum(const float* __restrict__ ex, const float* __restrict__ den,
                           const float* __restrict__ vn, const float* __restrict__ eatt,
                           const int* __restrict__ src, const int* __restrict__ dst,
                           float* __restrict__ attn) {
  long t = (long)blockIdx.x * blockDim.x + threadIdx.x;
  if (t >= (long)Ee * Dd) return;
  const int e = (int)(t >> 7), c = (int)(t & 127), h = c >> 5;
  const int s = src[e], d = dst[e];
  const float w = ex[(long)e * Hh + h] / den[(long)d * Hh + h];
  atomicAdd(&attn[(long)d * Dd + c], w * (vn[(long)s * Dd + c] + eatt[t]));
}

// ---------------- edge MLP epilogue ----------------
__global__ void bias_gelu(float* __restrict__ x, const float* __restrict__ b, int rows) {
  long i = (long)blockIdx.x * blockDim.x + threadIdx.x;
  if (i >= (long)rows * Dd) return;
  float v = x[i] + b[i & (Dd - 1)];
  x[i] = 0.5f * v * (1.f + erff(v * 0.7071067811865476f));  // exact gelu
}

// ---------------- context gate ----------------
__global__ void count_k(const int* __restrict__ batch, float* __restrict__ cnt) {
  int i = blockIdx.x * blockDim.x + threadIdx.x;
  if (i < Nn) atomicAdd(&cnt[batch[i]], 1.f);
}
__global__ void scatter_cv(const float* __restrict__ hv, const int* __restrict__ batch,
                           float* __restrict__ cV) {
  long t = (long)blockIdx.x * blockDim.x + threadIdx.x;
  if (t >= (long)Nn * Dd) return;
  atomicAdd(&cV[(long)batch[t >> 7] * Dd + (t & 127)], hv[t]);
}
__global__ void ctx_mlp1(const float* __restrict__ cV, const float* __restrict__ cnt,
                         const float* __restrict__ W, const float* __restrict__ b,
                         float* __restrict__ out) {
  const int r = blockIdx.x, c = threadIdx.x;
  const float inv = 1.f / fmaxf(cnt[r], 1.f);
  float s = 0.f;
  for (int j = 0; j < Dd; ++j) s += cV[r * Dd + j] * inv * W[j * Dd + c];
  out[r * Dd + c] = fmaxf(s + b[c], 0.f);
}
__global__ void ctx_mlp2(const float* __restrict__ in, const float* __restrict__ W,
                         const float* __restrict__ b, float* __restrict__ out) {
  const int r = blockIdx.x, c = threadIdx.x;
  float s = 0.f;
  for (int j = 0; j < Dd; ++j) s += in[r * Dd + j] * W[j * Dd + c];
  out[r * Dd + c] = 1.f / (1.f + expf(-(s + b[c])));
}
__global__ void gate_k(float* __restrict__ hv, const int* __restrict__ batch,
                       const float* __restrict__ g) {
  long t = (long)blockIdx.x * blockDim.x + threadIdx.x;
  if (t >= (long)Nn * Dd) return;
  hv[t] *= g[(long)batch[t >> 7] * Dd + (t & 127)];
}

// ---------------- host side ----------------
static inline long cdiv(long a, long b) { return (a + b - 1) / b; }

static void gemm(hipStream_t s, const float* A, const float* B, const float* bias,
                 float* C, const int* gather, int M, int K, int Nc, int accum, int act) {
  dim3 grid((unsigned)cdiv(M / 16, 4), (unsigned)(Nc / 16));
  gemm_f32_wmma<<<grid, 128, 0, s>>>(A, B, bias, C, gather, M, K, Nc, accum, act);
}
static void zero(hipStream_t s, float* p, long n) {
  zero_f<<<(unsigned)cdiv(n, 256), 256, 0, s>>>(p, n);
}

extern "C" void kernel_launch(void* const* d_in, const int* in_sizes, int n_in,
                              void* d_out, int out_size, void* d_ws, size_t ws_size,
                              hipStream_t stream) {
  (void)in_sizes; (void)n_in; (void)out_size; (void)ws_size;

  // inputs (jax pytree flatten: dict keys sorted, ASCII order)
  const float* hV   = (const float*)d_in[0];
  const int*   ei   = (const int*)d_in[1];
  const float* hE   = (const float*)d_in[2];
  const int*   batch= (const int*)d_in[4];
  const int* src = ei;        // edge_index[0]
  const int* dst = ei + Ee;   // edge_index[1]

  const float* We_W  = (const float*)d_in[5];
  const float* We_b  = (const float*)d_in[6];
  const float* Wv_W  = (const float*)d_in[7];
  const float* Wv_b  = (const float*)d_in[8];
  const float* bn_e_b= (const float*)d_in[9];
  const float* bn_e_g= (const float*)d_in[10];
  const float* bn_n_b= (const float*)d_in[11];
  const float* bn_n_g= (const float*)d_in[12];
  const float* ee_W  = (const float*)d_in[13];
  const float* ee_b  = (const float*)d_in[14];
  const float* l_We  = (const float*)d_in[15];
  const float* l_Wk  = (const float*)d_in[16];
  const float* l_Wq  = (const float*)d_in[17];
  const float* l_Wv  = (const float*)d_in[18];
  const float* l_bk  = (const float*)d_in[19];
  const float* l_bq  = (const float*)d_in[20];
  const float* l_bv  = (const float*)d_in[21];
  const float* l_cW1 = (const float*)d_in[22];
  const float* l_cW2 = (const float*)d_in[23];
  const float* l_cb1 = (const float*)d_in[24];
  const float* l_cb2 = (const float*)d_in[25];
  const float* l_eW11= (const float*)d_in[26];
  const float* l_eW12= (const float*)d_in[27];
  const float* l_eb11= (const float*)d_in[28];
  const float* l_eb12= (const float*)d_in[29];
  const float* l_ebnb= (const float*)d_in[30];
  const float* l_ebng= (const float*)d_in[31];
  const float* l_fW1 = (const float*)d_in[32];
  const float* l_fW2 = (const float*)d_in[33];
  const float* l_fb1 = (const float*)d_in[34];
  const float* l_fb2 = (const float*)d_in[35];
  const float* l_ln1b= (const float*)d_in[36];
  const float* l_ln1g= (const float*)d_in[37];
  const float* l_ln2b= (const float*)d_in[38];
  const float* l_ln2g= (const float*)d_in[39];
  const float* ne_W  = (const float*)d_in[40];
  const float* ne_b  = (const float*)d_in[41];

  // workspace carve-up
  float* w = (float*)d_ws;
  float* nb0   = w; w += (long)Nn * Dd;   // attn accumulator / ffn out scratch
  float* nb1   = w; w += (long)Nn * Dd;   // hv (persistent)
  float* qn    = w; w += (long)Nn * Dd;
  float* kn    = w; w += (long)Nn * Dd;
  float* vn    = w; w += (long)Nn * Dd;
  float* ffmid = w; w += (long)Nn * 512;
  float* eb0   = w; w += (long)Ee * Dd;
  float* eb1   = w; w += (long)Ee * Dd;
  float* eb2   = w; w += (long)Ee * Dd;
  float* logits= w; w += (long)Ee * Hh;
  float* ex    = w; w += (long)Ee * Hh;
  float* mxf   = w; w += (long)Nn * Hh;   // used as unsigned
  float* den   = w; w += (long)Nn * Hh;
  float* cV    = w; w += (long)Bb * Dd;
  float* gbuf  = w; w += (long)Bb * Dd;
  float* ctmp  = w; w += (long)Bb * Dd;
  float* stats = w; w += 256;
  float* cnt   = w; w += Bb;
  unsigned* mx = (unsigned*)mxf;

  // ---- graph counts ----
  zero(stream, cnt, Bb);
  count_k<<<(unsigned)cdiv(Nn, 256), 256, 0, stream>>>(batch, cnt);

  // ---- node embedding: hv = bn(hV@ne_W + ne_b) @ Wv_W + Wv_b ----
  gemm(stream, hV, ne_W, ne_b, nb0, nullptr, Nn, 1024, Dd, 0, 0);
  zero(stream, stats, 256);
  bn_stats<<<(unsigned)cdiv(Nn, 128), 128, 0, stream>>>(nb0, nullptr, nullptr, stats, Nn);
  bn_apply<<<(unsigned)cdiv((long)Nn * Dd, 256), 256, 0, stream>>>(
      nb0, nullptr, nullptr, stats, bn_n_g, bn_n_b, nb0, Nn);
  gemm(stream, nb0, Wv_W, Wv_b, nb1, nullptr, Nn, Dd, Dd, 0, 0);
  float* hv = nb1;

  // ---- edge embedding: he = bn(hE@ee_W + ee_b) @ We_W + We_b ----
  gemm(stream, hE, ee_W, ee_b, eb0, nullptr, Ee, 16, Dd, 0, 0);
  zero(stream, stats, 256);
  bn_stats<<<(unsigned)cdiv(Ee, 128), 128, 0, stream>>>(eb0, nullptr, nullptr, stats, Ee);
  bn_apply<<<(unsigned)cdiv((long)Ee * Dd, 256), 256, 0, stream>>>(
      eb0, nullptr, nullptr, stats, bn_e_g, bn_e_b, eb0, Ee);
  gemm(stream, eb0, We_W, We_b, eb1, nullptr, Ee, Dd, Dd, 0, 0);
  float* he = eb1; float* f0 = eb0; float* f1 = eb2;  // rotating edge buffers

  for (int l = 0; l < 4; ++l) {
    const long w128 = (long)l * Dd * Dd;
    const long b128 = (long)l * Dd;

    // --- TransformerConv ---
    gemm(stream, hv, l_Wq + w128, l_bq + b128, qn, nullptr, Nn, Dd, Dd, 0, 0);
    gemm(stream, hv, l_Wk + w128, l_bk + b128, kn, nullptr, Nn, Dd, Dd, 0, 0);
    gemm(stream, hv, l_Wv + w128, l_bv + b128, vn, nullptr, Nn, Dd, Dd, 0, 0);
    float* eatt = f0;
    gemm(stream, he, l_We + w128, nullptr, eatt, nullptr, Ee, Dd, Dd, 0, 0);

    init_mx<<<(unsigned)cdiv((long)Nn * Hh, 256), 256, 0, stream>>>(mx, (long)Nn * Hh);
    attn_logits<<<(unsigned)cdiv((long)Ee * Hh, 256), 256, 0, stream>>>(
        qn, kn, eatt, src, dst, logits, mx);
    zero(stream, den, (long)Nn * Hh);
    attn_exden<<<(unsigned)cdiv((long)Ee * Hh, 256), 256, 0, stream>>>(
        logits, dst, mx, ex, den);
    zero(stream, nb0, (long)Nn * Dd);
    attn_accum<<<(unsigned)cdiv((long)Ee * Dd, 256), 256, 0, stream>>>(
        ex, den, vn, eatt, src, dst, nb0);
    ln_res<<<(unsigned)cdiv(Nn, 8), 256, 0, stream>>>(hv, nb0, l_ln1g + b128, l_ln1b + b128, Nn);

    // --- FFN ---
    gemm(stream, hv, l_fW1 + (long)l * Dd * 512, l_fb1 + (long)l * 512, ffmid,
         nullptr, Nn, Dd, 512, 0, /*relu*/1);
    gemm(stream, ffmid, l_fW2 + (long)l * 512 * Dd, l_fb2 + b128, nb0,
         nullptr, Nn, 512, Dd, 0, 0);
    ln_res<<<(unsigned)cdiv(Nn, 8), 256, 0, stream>>>(hv, nb0, l_ln2g + b128, l_ln2b + b128, Nn);

    // --- EdgeMLP: msg = gelu([hv[src], he, hv[dst]] @ W11 + b11) @ W12 + b12 ---
    const float* W11 = l_eW11 + (long)l * 3 * Dd * Dd;  // [384,128] row-major
    float* msgA = f1;
    gemm(stream, hv, W11 + 0 * Dd * Dd, nullptr, msgA, src, Ee, Dd, Dd, 0, 0);
    gemm(stream, he, W11 + 1 * Dd * Dd, nullptr, msgA, nullptr, Ee, Dd, Dd, 1, 0);
    gemm(stream, hv, W11 + 2 * Dd * Dd, nullptr, msgA, dst, Ee, Dd, Dd, 1, 0);
    bias_gelu<<<(unsigned)cdiv((long)Ee * Dd, 256), 256, 0, stream>>>(msgA, l_eb11 + b128, Ee);
    float* msgB = f0;  // eatt buffer is dead now
    gemm(stream, msgA, l_eW12 + w128, nullptr, msgB, nullptr, Ee, Dd, Dd, 0, 0);
    // he_new = bn(he + msgB + e_b12)
    zero(stream, stats, 256);
    bn_stats<<<(unsigned)cdiv(Ee, 128), 128, 0, stream>>>(he, msgB, l_eb12 + b128, stats, Ee);
    bn_apply<<<(unsigned)cdiv((long)Ee * Dd, 256), 256, 0, stream>>>(
        he, msgB, l_eb12 + b128, stats, l_ebng + b128, l_ebnb + b128, msgA, Ee);
    { float* t = he; he = msgA; f1 = f0; f0 = t; }  // rotate: msgA -> he

    // --- context gate ---
    zero(stream, cV, (long)Bb * Dd);
    scatter_cv<<<(unsigned)cdiv((long)Nn * Dd, 256), 256, 0, stream>>>(hv, batch, cV);
    ctx_mlp1<<<Bb, Dd, 0, stream>>>(cV, cnt, l_cW1 + w128, l_cb1 + b128, ctmp);
    ctx_mlp2<<<Bb, Dd, 0, stream>>>(ctmp, l_cW2 + w128, l_cb2 + b128, gbuf);
    gate_k<<<(unsigned)cdiv((long)Nn * Dd, 256), 256, 0, stream>>>(hv, batch, gbuf);
  }

  hipMemcpyAsync(d_out, hv, (long)Nn * Dd * sizeof(float),
                 hipMemcpyDeviceToDevice, stream);
}